// dtcwtLayer_70437463654712
// MI455X (gfx1250) — compile-verified
//
#include <hip/hip_runtime.h>
#include <cstddef>

typedef float v2f __attribute__((ext_vector_type(2)));
typedef float v8f __attribute__((ext_vector_type(8)));

// -------- helpers (branchless) --------
__device__ __forceinline__ int symref(int i, int n) {
  i = (i < 0) ? (-1 - i) : i;
  i = (i >= n) ? (2 * n - 1 - i) : i;
  return i;
}

// 5-tap H0O, branchless select chain -> v_cndmask, no control flow
__device__ __forceinline__ float coefH0O(int t) {
  float r = 0.0f;
  r = (t == 0) ? -0.05f : r;
  r = (t == 1) ? 0.25f : r;
  r = (t == 2) ? 0.6f : r;
  r = (t == 3) ? 0.25f : r;
  r = (t == 4) ? -0.05f : r;
  return r;
}
// 7-tap H1O
__device__ __forceinline__ float coefH1O(int t) {
  float r = 0.0f;
  r = (t == 0) ? -0.010714285714285714f : r;
  r = (t == 1) ? 0.05357142857142857f : r;
  r = (t == 2) ? 0.26071428571428573f : r;
  r = (t == 3) ? -0.6071428571428571f : r;
  r = (t == 4) ? 0.26071428571428573f : r;
  r = (t == 5) ? 0.05357142857142857f : r;
  r = (t == 6) ? -0.010714285714285714f : r;
  return r;
}

// q-shift filters (10 taps) -- indexed with compile-time constants only
constexpr float H0A_c[10] = {0.03516384f, 0.0f, -0.08832942f, 0.23389032f, 0.76027237f,
                             0.5875183f, 0.0f, -0.11430184f, 0.0f, 0.0f};
constexpr float H0B_c[10] = {0.0f, 0.0f, -0.11430184f, 0.0f, 0.5875183f,
                             0.76027237f, 0.23389032f, -0.08832942f, 0.0f, 0.03516384f};
constexpr float H1A_c[10] = {0.0f, 0.0f, -0.11430184f, 0.0f, 0.5875183f,
                             -0.76027237f, 0.23389032f, 0.08832942f, 0.0f, -0.03516384f};
constexpr float H1B_c[10] = {-0.03516384f, 0.0f, 0.08832942f, 0.23389032f, -0.76027237f,
                             0.5875183f, 0.0f, -0.11430184f, 0.0f, 0.0f};

#define SQRT_HALF 0.70710678118654752f

// =====================================================================
// Kernel 1: fused level-1.  Per workgroup: one (b,ch) and a 32x32 tile.
//  vertical 5/7-tap colfilter  -> WMMA f32 16x16x4 chains (D = F * X)
//  horizontal colfilter        -> WMMA f32 16x16x4 chains (D = X * G)
//  writes lolo (full res, ws) and high0 (q2c bands, d_out)
// All wave-job maps are exactly 16 jobs / 8 waves x 2 iters: no divergent
// guards anywhere near WMMA (EXEC stays all-ones).
// =====================================================================
__global__ __launch_bounds__(256) void dtcwt_l1_wmma(const float* __restrict__ x,
                                                     float* __restrict__ lolo,
                                                     float* __restrict__ high0) {
  __shared__ float xs[40][64];        // input tile + halos; cols 38..63 / rows 38..39 zero
  __shared__ float slo[32][64];       // vertically lowpassed
  __shared__ float shi[32][64];       // vertically highpassed
  __shared__ float obuf[4][32][32];   // 0:lolo 1:hi*h0 2:lo*h1 3:hi*h1

  const int tid  = threadIdx.x;
  const int wave = tid >> 5;
  const int lane = tid & 31;
  const int M    = lane & 15;         // A: row  / B,C,D: column (N)
  const int hl   = lane >> 4;         // half-wave selector
  const int K0   = hl * 2;            // K offset within each K=4 chunk

  const int c0 = blockIdx.x * 32;
  const int r0 = blockIdx.y * 32;
  const int b  = blockIdx.z / 3;
  const int ch = blockIdx.z % 3;

  // Per-lane band-matrix coefficients, computed ONCE (branchless selects).
  // cLo[2c],cLo[2c+1] = H0O band entry at (row M, K-col 4c+K0 / +1); same for cHi.
  float cLo[12], cHi[12];
#pragma unroll
  for (int c = 0; c < 6; ++c) {
    int j0 = 4 * c + K0;
    cLo[2 * c]     = coefH0O(j0 - M - 1);  // lo: off = 3-mh = 1
    cLo[2 * c + 1] = coefH0O(j0 - M);
    cHi[2 * c]     = coefH1O(j0 - M);      // hi: off = 0
    cHi[2 * c + 1] = coefH1O(j0 - M + 1);
  }

  // zero LDS tile (so padded rows/cols contribute exact zeros)
  for (int e = tid; e < 40 * 64; e += 256) (&xs[0][0])[e] = 0.0f;
  __syncthreads();
  // load 38x38 input window with symmetric reflection
  for (int e = tid; e < 38 * 38; e += 256) {
    int i = e / 38, j = e % 38;
    int rr = symref(r0 - 3 + i, 512);
    int cc = symref(c0 - 3 + j, 512);
    xs[i][j] = x[((size_t)((b * 512 + rr) * 512 + cc)) * 3 + ch];
  }
  __syncthreads();

  // ---- vertical stage: 16 jobs = {lo,hi} x 2 row-chunks x 4 col-groups
  for (int it = 0; it < 2; ++it) {
    int job = wave + 8 * it;
    int f = job >> 3, rc = (job >> 2) & 1, cg = job & 3;
    v8f acc = {0.f, 0.f, 0.f, 0.f, 0.f, 0.f, 0.f, 0.f};
#pragma unroll
    for (int c = 0; c < 6; ++c) {      // K = 24 in chunks of 4
      int j0 = 4 * c + K0, j1 = j0 + 1;
      v2f a, bm;
      a.x = (f == 0) ? cLo[2 * c] : cHi[2 * c];
      a.y = (f == 0) ? cLo[2 * c + 1] : cHi[2 * c + 1];
      bm.x = xs[rc * 16 + j0][cg * 16 + M];
      bm.y = xs[rc * 16 + j1][cg * 16 + M];
      acc = __builtin_amdgcn_wmma_f32_16x16x4_f32(false, a, false, bm, (short)0, acc,
                                                  false, false);
    }
    float (*S)[64] = (f == 0) ? slo : shi;
#pragma unroll
    for (int v = 0; v < 8; ++v) S[rc * 16 + v + 8 * hl][cg * 16 + M] = acc[v];
  }
  __syncthreads();

  // ---- horizontal stage: 16 jobs = 4 outputs x 2 row-chunks x 2 col-chunks
  for (int it = 0; it < 2; ++it) {
    int job = wave + 8 * it;
    int outsel = job >> 2, rc = (job >> 1) & 1, cc = job & 1;
    float (*S)[64] = (outsel == 0 || outsel == 2) ? slo : shi;
    int fh = (outsel <= 1) ? 0 : 1;
    v8f acc = {0.f, 0.f, 0.f, 0.f, 0.f, 0.f, 0.f, 0.f};
#pragma unroll
    for (int c = 0; c < 6; ++c) {
      int j0 = 4 * c + K0, j1 = j0 + 1;
      v2f a, bm;
      a.x = S[rc * 16 + M][cc * 16 + j0];
      a.y = S[rc * 16 + M][cc * 16 + j1];
      bm.x = (fh == 0) ? cLo[2 * c] : cHi[2 * c];
      bm.y = (fh == 0) ? cLo[2 * c + 1] : cHi[2 * c + 1];
      acc = __builtin_amdgcn_wmma_f32_16x16x4_f32(false, a, false, bm, (short)0, acc,
                                                  false, false);
    }
#pragma unroll
    for (int v = 0; v < 8; ++v) obuf[outsel][rc * 16 + v + 8 * hl][cc * 16 + M] = acc[v];
  }
  __syncthreads();

  // ---- write lolo (full resolution)
  for (int e = tid; e < 1024; e += 256) {
    int R = e >> 5, C = e & 31;
    lolo[((size_t)((b * 512 + r0 + R) * 512 + (c0 + C))) * 3 + ch] = obuf[0][R][C];
  }
  // ---- q2c + write high0: bands hi*h0->(0,5), lo*h1->(2,3), hi*h1->(1,4)
  {
    int i = tid >> 4, j = tid & 15;
    int I = (r0 >> 1) + i, J = (c0 >> 1) + j;
    size_t base = ((size_t)((b * 256 + I) * 256 + J)) * 36;
#pragma unroll
    for (int q = 0; q < 3; ++q) {
      int src = q + 1;
      int p = (q == 0) ? 0 : (q == 1) ? 2 : 1;
      int r = (q == 0) ? 5 : (q == 1) ? 3 : 4;
      float a_ = obuf[src][2 * i][2 * j];
      float b_ = obuf[src][2 * i][2 * j + 1];
      float c_ = obuf[src][2 * i + 1][2 * j];
      float d_ = obuf[src][2 * i + 1][2 * j + 1];
      high0[base + ch * 6 + p]      = (a_ - d_) * SQRT_HALF;
      high0[base + 18 + ch * 6 + p] = (b_ + c_) * SQRT_HALF;
      high0[base + ch * 6 + r]      = (a_ + d_) * SQRT_HALF;
      high0[base + 18 + ch * 6 + r] = (b_ - c_) * SQRT_HALF;
    }
  }
}

// =====================================================================
// Kernel 2: level-2 coldfilt along rows (512 -> 256).
//  out[2t]   (lo2) = sum_j H0B[j]*xe[4t+20-2j]   (even phase)
//  out[2t+1] (lo2) = sum_j H0A[j]*xe[4t+21-2j]   (odd phase)
//  hi2 swaps phases because sum(H1B*H1A) < 0.
// =====================================================================
__global__ __launch_bounds__(256) void dtcwt_l2_rows(const float* __restrict__ lolo,
                                                     float* __restrict__ lo2,
                                                     float* __restrict__ hi2) {
  size_t id = (size_t)blockIdx.x * 256 + threadIdx.x;  // 16*128*512*3 threads
  int ch = (int)(id % 3);
  size_t r = id / 3;
  int c = (int)(r % 512); r /= 512;
  int t = (int)(r % 128);
  int b = (int)(r / 128);

  float se_h0b = 0.f, so_h0a = 0.f, so_h1a = 0.f, se_h1b = 0.f;
#pragma unroll
  for (int k = 0; k < 10; ++k) {
    int re = symref(4 * t - 8 + 2 * k, 512);
    int ro = symref(4 * t - 7 + 2 * k, 512);
    float e = lolo[((size_t)((b * 512 + re) * 512 + c)) * 3 + ch];
    float o = lolo[((size_t)((b * 512 + ro) * 512 + c)) * 3 + ch];
    se_h0b += H0B_c[9 - k] * e;
    se_h1b += H1B_c[9 - k] * e;
    so_h0a += H0A_c[9 - k] * o;
    so_h1a += H1A_c[9 - k] * o;
  }
  size_t o0 = ((size_t)((b * 256 + 2 * t) * 512 + c)) * 3 + ch;
  size_t o1 = ((size_t)((b * 256 + 2 * t + 1) * 512 + c)) * 3 + ch;
  lo2[o0] = se_h0b;  lo2[o1] = so_h0a;
  hi2[o0] = so_h1a;  hi2[o1] = se_h1b;
}

// =====================================================================
// Kernel 3: level-2 coldfilt along cols + q2c -> lolo2 & high1.
// One thread per (b,i,j,ch); covers rows {2i,2i+1}, cols {2j,2j+1}.
// =====================================================================
__global__ __launch_bounds__(256) void dtcwt_l2_cols(const float* __restrict__ lo2,
                                                     const float* __restrict__ hi2,
                                                     float* __restrict__ lolo2,
                                                     float* __restrict__ high1) {
  size_t id = (size_t)blockIdx.x * 256 + threadIdx.x;  // 16*128*128*3 threads
  int ch = (int)(id % 3);
  size_t r = id / 3;
  int j = (int)(r % 128); r /= 128;
  int i = (int)(r % 128);
  int b = (int)(r / 128);

  float bh0[2][2], bl1[2][2], bh1[2][2];
#pragma unroll
  for (int dr = 0; dr < 2; ++dr) {
    int R = 2 * i + dr;
    float L0e = 0.f, L0o = 0.f, L1o = 0.f, L1e = 0.f;
    float G0e = 0.f, G0o = 0.f, G1o = 0.f, G1e = 0.f;
#pragma unroll
    for (int k = 0; k < 10; ++k) {
      int ce = symref(4 * j - 8 + 2 * k, 512);
      int co = symref(4 * j - 7 + 2 * k, 512);
      size_t be = ((size_t)((b * 256 + R) * 512 + ce)) * 3 + ch;
      size_t bo = ((size_t)((b * 256 + R) * 512 + co)) * 3 + ch;
      float le = lo2[be], lo_ = lo2[bo];
      float he = hi2[be], ho = hi2[bo];
      L0e += H0B_c[9 - k] * le;   L1e += H1B_c[9 - k] * le;
      L0o += H0A_c[9 - k] * lo_;  L1o += H1A_c[9 - k] * lo_;
      G0e += H0B_c[9 - k] * he;   G1e += H1B_c[9 - k] * he;
      G0o += H0A_c[9 - k] * ho;   G1o += H1A_c[9 - k] * ho;
    }
    lolo2[((size_t)((b * 256 + R) * 256 + 2 * j)) * 3 + ch]     = L0e;
    lolo2[((size_t)((b * 256 + R) * 256 + 2 * j + 1)) * 3 + ch] = L0o;
    bh0[dr][0] = G0e; bh0[dr][1] = G0o;   // H0 on hi2: even, odd phases
    bl1[dr][0] = L1o; bl1[dr][1] = L1e;   // H1 on lo2: odd first (sum<0)
    bh1[dr][0] = G1o; bh1[dr][1] = G1e;   // H1 on hi2: odd first
  }
  size_t base = ((size_t)((b * 128 + i) * 128 + j)) * 36;
  auto emit = [&](float (&B)[2][2], int p, int q) {
    float a_ = B[0][0], b_ = B[0][1], c_ = B[1][0], d_ = B[1][1];
    high1[base + p * 3 + ch]      = (a_ - d_) * SQRT_HALF;
    high1[base + 18 + p * 3 + ch] = (b_ + c_) * SQRT_HALF;
    high1[base + q * 3 + ch]      = (a_ + d_) * SQRT_HALF;
    high1[base + 18 + q * 3 + ch] = (b_ - c_) * SQRT_HALF;
  };
  emit(bh0, 0, 5);
  emit(bl1, 2, 3);
  emit(bh1, 1, 4);
}

// =====================================================================
extern "C" void kernel_launch(void* const* d_in, const int* in_sizes, int n_in,
                              void* d_out, int out_size, void* d_ws, size_t ws_size,
                              hipStream_t stream) {
  (void)in_sizes; (void)n_in; (void)out_size; (void)ws_size;
  const float* x = (const float*)d_in[0];
  float* out = (float*)d_out;

  // workspace layout (floats): lolo (16*512*512*3) | lo2 (16*256*512*3) | hi2
  float* lolo = (float*)d_ws;
  float* lo2  = lolo + 12582912;
  float* hi2  = lo2 + 6291456;

  // output layout (floats): lolo2 | high0 | high1
  float* lolo2 = out;
  float* high0 = out + 3145728;    // 16*256*256*3
  float* high1 = out + 40894464;   // + 16*256*256*36

  dim3 g1(16, 16, 48);
  dtcwt_l1_wmma<<<g1, 256, 0, stream>>>(x, lolo, high0);
  dtcwt_l2_rows<<<12288, 256, 0, stream>>>(lolo, lo2, hi2);       // 16*128*512*3 / 256
  dtcwt_l2_cols<<<3072, 256, 0, stream>>>(lo2, hi2, lolo2, high1); // 16*128*128*3 / 256
}